// CountVectorizer_31404800868363
// MI455X (gfx1250) — compile-verified
//
#include <hip/hip_runtime.h>
#include <stdint.h>

// CountVectorizer hash + probe + scatter for MI455X (gfx1250, wave32).
// No matrix ops exist in this workload; the MI455X-relevant levers are
// NT-streamed b128 loads, L2-resident gather tables, and no-return
// global_atomic_add_f32 into an L2-resident output.

#define VOCAB 32000u   // V (fixed by the reference problem)
#define WGROUPS 4      // W packed 4-byte groups per word
#define COLL 2         // C collision slots

// native clang vector types (required by __builtin_nontemporal_load)
typedef int   v4i __attribute__((ext_vector_type(4)));
typedef int   v2i __attribute__((ext_vector_type(2)));
typedef float v4f __attribute__((ext_vector_type(4)));

// fused murmur constant: C1*C2 mod 2^32
static constexpr uint32_t kCM = (uint32_t)(3432918353u * 461845907u);

__global__ __launch_bounds__(256) void cv_zero_kernel(float* __restrict__ out, int n)
{
    int i = blockIdx.x * blockDim.x + threadIdx.x;
    int i4 = i * 4;
    if (i4 + 3 < n) {
        v4f z = {0.f, 0.f, 0.f, 0.f};
        *reinterpret_cast<v4f*>(out + i4) = z;
    } else {
        for (int j = i4; j < n; ++j) out[j] = 0.f;
    }
}

__device__ __forceinline__ uint32_t pack4(const v4i b)
{
    // bytes are in [0,128): identical to sum(b * [1,2^8,2^16,2^24])
    return (uint32_t)b.x | ((uint32_t)b.y << 8) |
           ((uint32_t)b.z << 16) | ((uint32_t)b.w << 24);
}

__global__ __launch_bounds__(256) void cv_hash_count_kernel(
    const v4i* __restrict__ docs,       // [N, 4] v4i  (== [N, W=4, 4] int32 bytes)
    const int* __restrict__ doc_ids,    // [N]
    const v4i* __restrict__ table,      // [V, 2] v4i  (== [V, C=2, W=4] int32)
    const v2i* __restrict__ fidx,       // [V] v2i     (== [V, C=2] int32)
    const float* __restrict__ scale_p,  // [1]
    float* __restrict__ out,            // [BATCH, V]
    int N)
{
    int t = blockIdx.x * blockDim.x + threadIdx.x;
    if (t >= N) return;

    // --- stream the word's 16 bytes (64B) with non-temporal b128 loads ---
    const v4i* dp = docs + (size_t)t * WGROUPS;
    v4i b0 = __builtin_nontemporal_load(dp + 0);
    v4i b1 = __builtin_nontemporal_load(dp + 1);
    v4i b2 = __builtin_nontemporal_load(dp + 2);
    v4i b3 = __builtin_nontemporal_load(dp + 3);

    uint32_t p0 = pack4(b0);
    uint32_t p1 = pack4(b1);
    uint32_t p2 = pack4(b2);
    uint32_t p3 = pack4(b3);

    // --- rolling hash: h = XOR_j (p_j * C1 * C2)  (C1*C2 pre-fused mod 2^32) ---
    uint32_t h = (p0 * kCM) ^ (p1 * kCM) ^ (p2 * kCM) ^ (p3 * kCM);
    uint32_t idx = h % VOCAB;   // compile-time divisor -> magic-multiply codegen

    // --- probe the L2-resident collision table (2 x b128 + 1 x b64) ---
    const v4i* te = table + (size_t)idx * COLL;
    v4i e0 = te[0];
    v4i e1 = te[1];
    v2i fi = fidx[idx];

    bool m0 = (e0.x == (int)p0) & (e0.y == (int)p1) &
              (e0.z == (int)p2) & (e0.w == (int)p3);
    bool m1 = (e1.x == (int)p0) & (e1.y == (int)p1) &
              (e1.z == (int)p2) & (e1.w == (int)p3);

    int feat = (m0 ? fi.x : 0) + (m1 ? fi.y : 0);

    if (feat != 0) {
        int d = doc_ids[t];
        float s = scale_p[0];
        // no-return global_atomic_add_f32; output fits in 192MB L2.
        atomicAdd(out + (size_t)d * (size_t)VOCAB + (uint32_t)(feat - 1), s);
    }
}

extern "C" void kernel_launch(void* const* d_in, const int* in_sizes, int n_in,
                              void* d_out, int out_size, void* d_ws, size_t ws_size,
                              hipStream_t stream)
{
    (void)n_in; (void)d_ws; (void)ws_size;

    const v4i*   docs    = (const v4i*)d_in[0];    // documents [N, W, 4] int32
    const int*   doc_ids = (const int*)d_in[1];    // [N]
    // d_in[2] = batch_size scalar (unused; out_size encodes BATCH*V)
    const v4i*   table   = (const v4i*)d_in[3];    // hash_table [V, C, W] int32
    const v2i*   fidx    = (const v2i*)d_in[4];    // feature_indices [V, C] int32
    // d_in[5] = powers_of_two (compile-time constant packing, unused)
    const float* scale   = (const float*)d_in[6];  // [1]
    float*       out     = (float*)d_out;

    const int N = in_sizes[1];                     // words = len(doc_ids)

    // 1) zero the output (harness poisons d_out with 0xAA)
    {
        int n4 = (out_size + 3) / 4;
        int blocks = (n4 + 255) / 256;
        cv_zero_kernel<<<blocks, 256, 0, stream>>>(out, out_size);
    }

    // 2) hash + probe + scatter-add
    {
        int blocks = (N + 255) / 256;
        cv_hash_count_kernel<<<blocks, 256, 0, stream>>>(
            docs, doc_ids, table, fidx, scale, out, N);
    }
}